// EdgeConvNoC_41257455845942
// MI455X (gfx1250) — compile-verified
//
#include <hip/hip_runtime.h>

#define B_    4
#define CIN   64
#define COUT  64
#define NPTS  16384
#define KNN   16
#define EPSV  1e-5f

typedef __attribute__((ext_vector_type(2))) float v2f;
typedef __attribute__((ext_vector_type(8))) float v8f;

// ---------------------------------------------------------------------------
// Kernel 1: fused dual GEMM  local = w1*F, edge = w2*F   (per batch)
// One wave computes a 64(Cout) x 16(points) tile for BOTH weight matrices
// using V_WMMA_F32_16X16X4_F32, K-loop of 16 steps over C_IN=64.
// Results stored point-major [b][n][o] so the later gather reads contiguous
// 256B columns.
// ---------------------------------------------------------------------------
__global__ __launch_bounds__(256) void gemm_wmma(const float* __restrict__ F,
                                                 const float* __restrict__ w1,
                                                 const float* __restrict__ w2,
                                                 float* __restrict__ localT,
                                                 float* __restrict__ edgeT) {
    int tid  = threadIdx.x;
    int wave = (blockIdx.x * 256 + tid) >> 5;   // 4096 waves total
    int lane = tid & 31;
    int b    = wave >> 10;                      // NPTS/16 = 1024 n-tiles per batch
    int n0   = (wave & 1023) << 4;

    const float* Fb = F + (size_t)b * CIN * NPTS;
    int col   = n0 + (lane & 15);
    int khalf = (lane >> 4) << 1;               // 0 or 2 (K sub-column per lane half)

    v8f acc[8] = {};                            // [w1 m0..m3 | w2 m0..m3]

    #pragma unroll
    for (int t = 0; t < 16; ++t) {
        int krow = t * 4 + khalf;
        // B fragment (4x16 of feature): VGPR0 = rows kk+0/kk+2, VGPR1 = kk+1/kk+3
        v2f bf;
        bf.x = Fb[(size_t)krow       * NPTS + col];
        bf.y = Fb[(size_t)(krow + 1) * NPTS + col];
        #pragma unroll
        for (int m = 0; m < 4; ++m) {
            int widx = (m * 16 + (lane & 15)) * CIN + krow;  // 8B aligned (krow even)
            v2f a1 = *(const v2f*)(w1 + widx);
            v2f a2 = *(const v2f*)(w2 + widx);
            acc[m]     = __builtin_amdgcn_wmma_f32_16x16x4_f32(
                             false, a1, false, bf, (short)0, acc[m],     false, false);
            acc[4 + m] = __builtin_amdgcn_wmma_f32_16x16x4_f32(
                             false, a2, false, bf, (short)0, acc[4 + m], false, false);
        }
    }

    // D layout: VGPR i -> o = 16m + 8*(lane>>4) + i, n = n0 + (lane&15).
    // Per lane the 8 accum elements are 8 consecutive channels -> two float4 stores.
    size_t pbase = ((size_t)b * NPTS + n0 + (lane & 15)) * COUT + ((lane >> 4) << 3);
    #pragma unroll
    for (int m = 0; m < 4; ++m) {
        float4 l0 = make_float4(acc[m][0], acc[m][1], acc[m][2], acc[m][3]);
        float4 l1 = make_float4(acc[m][4], acc[m][5], acc[m][6], acc[m][7]);
        float4 e0 = make_float4(acc[4+m][0], acc[4+m][1], acc[4+m][2], acc[4+m][3]);
        float4 e1 = make_float4(acc[4+m][4], acc[4+m][5], acc[4+m][6], acc[4+m][7]);
        *(float4*)(localT + pbase + m * 16)     = l0;
        *(float4*)(localT + pbase + m * 16 + 4) = l1;
        *(float4*)(edgeT  + pbase + m * 16)     = e0;
        *(float4*)(edgeT  + pbase + m * 16 + 4) = e1;
    }
}

// ---------------------------------------------------------------------------
// Kernel 2: per-channel partial sum / sumsq of diff = edge[gather] - local.
// 64-thread channel groups; each block handles 256 points; deterministic
// per-block partials written to workspace (no float atomics).
// ---------------------------------------------------------------------------
__global__ __launch_bounds__(256) void stats_partial(const float* __restrict__ localT,
                                                     const float* __restrict__ edgeT,
                                                     const int* __restrict__ knn,
                                                     float* __restrict__ partial) {
    __shared__ float ls[256];
    __shared__ float lq[256];
    int tid = threadIdx.x;
    int o   = tid & 63;
    int grp = tid >> 6;
    float s = 0.f, q = 0.f;
    int pEnd = (blockIdx.x + 1) * 256;
    for (int p = blockIdx.x * 256 + grp; p < pEnd; p += 4) {
        int b = p >> 14;                         // NPTS = 2^14
        float l = localT[(size_t)p * COUT + o];
        const int* idx = knn + (size_t)p * KNN;
        size_t ebase = (size_t)b * NPTS * COUT;
        #pragma unroll
        for (int k = 0; k < KNN; ++k) {
            int j = idx[k];
            float d = edgeT[ebase + (size_t)j * COUT + o] - l;
            s += d;
            q += d * d;
        }
    }
    ls[tid] = s; lq[tid] = q;
    __syncthreads();
    if (tid < 64) {
        float ts = ls[tid] + ls[tid + 64] + ls[tid + 128] + ls[tid + 192];
        float tq = lq[tid] + lq[tid + 64] + lq[tid + 128] + lq[tid + 192];
        partial[(size_t)blockIdx.x * 128 + tid]      = ts;
        partial[(size_t)blockIdx.x * 128 + 64 + tid] = tq;
    }
}

// ---------------------------------------------------------------------------
// Kernel 3: fixed-order reduction of partials + fold mean/var/gamma/beta into
// per-channel affine coeffs:  normed = d*a + c,  a = gamma*rsqrt(var+eps),
// c = beta - mu*a.
// ---------------------------------------------------------------------------
__global__ void finalize_stats(const float* __restrict__ partial,
                               const float* __restrict__ gamma,
                               const float* __restrict__ beta,
                               float* __restrict__ coeff) {
    int o = threadIdx.x;                        // 64 threads
    float s = 0.f, q = 0.f;
    for (int blk = 0; blk < 256; ++blk) {
        s += partial[blk * 128 + o];
        q += partial[blk * 128 + 64 + o];
    }
    const float inv = 1.0f / (float)((size_t)B_ * NPTS * KNN);
    float mu  = s * inv;
    float var = q * inv - mu * mu;
    float r   = rsqrtf(var + EPSV);
    float a   = gamma[o] * r;
    float c   = beta[o] - mu * a;
    coeff[o]      = a;
    coeff[64 + o] = c;
}

// ---------------------------------------------------------------------------
// Kernel 4: re-gather (L2-resident), apply affine + ReLU, mean over K,
// write channel-major output (B, C_OUT, N).
// ---------------------------------------------------------------------------
__global__ __launch_bounds__(256) void apply_norm(const float* __restrict__ localT,
                                                  const float* __restrict__ edgeT,
                                                  const int* __restrict__ knn,
                                                  const float* __restrict__ coeff,
                                                  float* __restrict__ out) {
    int tid = threadIdx.x;
    int o   = tid & 63;
    int grp = tid >> 6;
    float a = coeff[o];
    float c = coeff[64 + o];
    int pEnd = (blockIdx.x + 1) * 64;
    for (int p = blockIdx.x * 64 + grp; p < pEnd; p += 4) {
        int b = p >> 14;
        int n = p & (NPTS - 1);
        float l = localT[(size_t)p * COUT + o];
        const int* idx = knn + (size_t)p * KNN;
        size_t ebase = (size_t)b * NPTS * COUT;
        float acc = 0.f;
        #pragma unroll
        for (int k = 0; k < KNN; ++k) {
            int j = idx[k];
            float d = edgeT[ebase + (size_t)j * COUT + o] - l;
            float v = fmaf(d, a, c);
            acc += fmaxf(v, 0.0f);
        }
        out[((size_t)b * COUT + o) * NPTS + n] = acc * (1.0f / KNN);
    }
}

// ---------------------------------------------------------------------------
extern "C" void kernel_launch(void* const* d_in, const int* in_sizes, int n_in,
                              void* d_out, int out_size, void* d_ws, size_t ws_size,
                              hipStream_t stream) {
    const float* feature = (const float*)d_in[0];
    const float* w1      = (const float*)d_in[1];
    const float* w2      = (const float*)d_in[2];
    const float* gamma   = (const float*)d_in[3];
    const float* beta    = (const float*)d_in[4];
    const int*   knn     = (const int*)d_in[5];
    float* out = (float*)d_out;

    float* ws      = (float*)d_ws;
    float* localT  = ws;                                  // B*N*64 floats (16 MB)
    float* edgeT   = localT + (size_t)B_ * NPTS * COUT;   // B*N*64 floats (16 MB)
    float* partial = edgeT  + (size_t)B_ * NPTS * COUT;   // 256 blocks * 128 floats
    float* coeff   = partial + 256 * 128;                 // 128 floats

    gemm_wmma<<<512, 256, 0, stream>>>(feature, w1, w2, localT, edgeT);
    stats_partial<<<256, 256, 0, stream>>>(localT, edgeT, knn, partial);
    finalize_stats<<<1, 64, 0, stream>>>(partial, gamma, beta, coeff);
    apply_norm<<<1024, 256, 0, stream>>>(localT, edgeT, knn, coeff, out);
}